// MyLoss_53257594470622
// MI455X (gfx1250) — compile-verified
//
#include <hip/hip_runtime.h>

#define TW 512
#define TH 512
#define TB 64
constexpr float IRE = 1.0f / 400.0f;

typedef float v2f __attribute__((ext_vector_type(2)));
typedef float v8f __attribute__((ext_vector_type(8)));

// Exact 32-lane sum. WMMA path: D = ones(16x4) * B(4x16) sums along K, so the
// 32 per-lane values (in B's first VGPR; second VGPR = 0) collapse into 16
// column sums replicated over D's rows; 4 xor-shuffles finish the 16-column sum.
__device__ __forceinline__ float wave_sum32(float x) {
#if __has_builtin(__builtin_amdgcn_wmma_f32_16x16x4_f32)
  v2f a; a[0] = 1.0f; a[1] = 1.0f;   // A = ones (16x4 f32)
  v2f b; b[0] = x;    b[1] = 0.0f;   // B carries the 32 lane partials
  v8f c = {};
  c = __builtin_amdgcn_wmma_f32_16x16x4_f32(
      /*neg_a=*/false, a, /*neg_b=*/false, b,
      /*c_mod=*/(short)0, c, /*reuse_a=*/false, /*reuse_b=*/false);
  float s = c[0];                    // lane l holds column (l%16) sum
#else
  float s = x;
  s += __shfl_xor(s, 16, 32);
#endif
  s += __shfl_xor(s, 1, 32);
  s += __shfl_xor(s, 2, 32);
  s += __shfl_xor(s, 4, 32);
  s += __shfl_xor(s, 8, 32);
  return s;                          // every lane: total over 32 lanes
}

struct Res { float mass, mu, mv; };

// l/r = +/-1 along axis2 (row, stride W); b/t = +/-1 along axis3 (contiguous)
__device__ __forceinline__ Res residuals(float um, float vm,
                                         float ul, float vl,
                                         float ur, float vr,
                                         float ub, float vb,
                                         float ut, float vt) {
  float dudx  = (ur - ul) * 0.5f;
  float dvdy  = (vt - vb) * 0.5f;
  float du2dx = 0.25f * ((ur + um) * (ur + um) - (ul + um) * (ul + um));
  float duvdy = 0.25f * ((ut + um) * (vt + vm) - (ub + um) * (vb + vm));
  float dv2dy = 0.25f * ((vt + vm) * (vt + vm) - (vb + vm) * (vb + vm));
  float dvudx = 0.25f * ((vr + vm) * (ur + um) - (vl + vm) * (ul + um));
  float dudx2 = ur - 2.0f * um + ul;
  float dudy2 = ut - 2.0f * um + ub;
  float dvdx2 = vr - 2.0f * vm + vl;
  float dvdy2 = vt - 2.0f * vm + vb;
  Res r;
  r.mass = dudx + dvdy;
  r.mu   = du2dx + duvdy - (dudx2 + dudy2) * IRE;
  r.mv   = dvudx + dv2dy - (dvdx2 + dvdy2) * IRE;
  return r;
}

// One block per (batch, row). Single fused pass: L1 over all points, stencil
// residual diffs over the interior. Deterministic per-block partial sums to ws.
__global__ void __launch_bounds__(256)
loss_partial_kernel(const float* __restrict__ pred,
                    const float* __restrict__ trut,
                    float* __restrict__ ws) {
  const int bi = blockIdx.x;
  const int b  = bi >> 9;      // / 512
  const int i  = bi & 511;     // row

  const size_t plane = (size_t)TH * TW;
  const size_t offU  = ((size_t)b * 2 + 0) * plane + (size_t)i * TW;
  const size_t offV  = ((size_t)b * 2 + 1) * plane + (size_t)i * TW;
  const float* pu = pred + offU;
  const float* pv = pred + offV;
  const float* tu = trut + offU;
  const float* tv = trut + offV;

  float s_mass = 0.0f, s_mom = 0.0f, s_l1 = 0.0f;
  const bool irow = (i >= 1) && (i <= TH - 2);

  for (int j = threadIdx.x; j < TW; j += 256) {
    float pum = pu[j], pvm = pv[j];
    float tum = tu[j], tvm = tv[j];
    s_l1 += fabsf(pum - tum) + fabsf(pvm - tvm);

    if (irow && j >= 1 && j <= TW - 2) {
      float pul = pu[j - TW], pur = pu[j + TW], pub = pu[j - 1], put = pu[j + 1];
      float pvl = pv[j - TW], pvr = pv[j + TW], pvb = pv[j - 1], pvt = pv[j + 1];
      float tul = tu[j - TW], tur = tu[j + TW], tub = tu[j - 1], tut = tu[j + 1];
      float tvl = tv[j - TW], tvr = tv[j + TW], tvb = tv[j - 1], tvt = tv[j + 1];
      Res rp = residuals(pum, pvm, pul, pvl, pur, pvr, pub, pvb, put, pvt);
      Res rt = residuals(tum, tvm, tul, tvl, tur, tvr, tub, tvb, tut, tvt);
      s_mass += fabsf(rt.mass - rp.mass);
      s_mom  += fabsf(rt.mu - rp.mu) + fabsf(rt.mv - rp.mv);
    }
  }

  // All 256 threads converged here -> EXEC all-1s for WMMA.
  float w0 = wave_sum32(s_mass);
  float w1 = wave_sum32(s_mom);
  float w2 = wave_sum32(s_l1);

  __shared__ float red[3][8];
  const int lane = threadIdx.x & 31;
  const int wave = threadIdx.x >> 5;
  if (lane == 0) { red[0][wave] = w0; red[1][wave] = w1; red[2][wave] = w2; }
  __syncthreads();
  if (threadIdx.x == 0) {
    float a = 0.0f, m = 0.0f, c = 0.0f;
    #pragma unroll
    for (int k = 0; k < 8; ++k) { a += red[0][k]; m += red[1][k]; c += red[2][k]; }
    ws[(size_t)bi * 3 + 0] = a;
    ws[(size_t)bi * 3 + 1] = m;
    ws[(size_t)bi * 3 + 2] = c;
  }
}

// Single-block deterministic final reduction in double precision.
__global__ void __launch_bounds__(256)
loss_final_kernel(const float* __restrict__ ws, int nblocks,
                  float* __restrict__ out) {
  double a = 0.0, m = 0.0, c = 0.0;
  for (int k = threadIdx.x; k < nblocks; k += 256) {
    a += (double)ws[(size_t)k * 3 + 0];
    m += (double)ws[(size_t)k * 3 + 1];
    c += (double)ws[(size_t)k * 3 + 2];
  }
  __shared__ double r[3][256];
  r[0][threadIdx.x] = a; r[1][threadIdx.x] = m; r[2][threadIdx.x] = c;
  __syncthreads();
  for (int s = 128; s > 0; s >>= 1) {
    if (threadIdx.x < (unsigned)s) {
      r[0][threadIdx.x] += r[0][threadIdx.x + s];
      r[1][threadIdx.x] += r[1][threadIdx.x + s];
      r[2][threadIdx.x] += r[2][threadIdx.x + s];
    }
    __syncthreads();
  }
  if (threadIdx.x == 0) {
    const double N1 = 64.0 * 510.0 * 510.0;            // interior points
    const double N2 = 64.0 * 2.0 * 512.0 * 512.0;      // all elements
    double loss = (5.0 * (r[0][0] / N1) + 25.0 * (r[1][0] / N1) + (r[2][0] / N2)) / 3.0;
    out[0] = (float)loss;
  }
}

extern "C" void kernel_launch(void* const* d_in, const int* in_sizes, int n_in,
                              void* d_out, int out_size, void* d_ws, size_t ws_size,
                              hipStream_t stream) {
  (void)in_sizes; (void)n_in; (void)out_size; (void)ws_size;
  const float* pred = (const float*)d_in[0];
  const float* trut = (const float*)d_in[1];
  float* ws  = (float*)d_ws;          // 64*512 blocks * 3 floats = 384 KB
  float* out = (float*)d_out;

  const int nblocks = TB * TH;        // 32768
  loss_partial_kernel<<<nblocks, 256, 0, stream>>>(pred, trut, ws);
  loss_final_kernel<<<1, 256, 0, stream>>>(ws, nblocks, out);
}